// GraphAttentionLayer_6107443494967
// MI455X (gfx1250) — compile-verified
//
#include <hip/hip_runtime.h>

// GAT layer, fused single kernel: one workgroup (256 thr / 8 waves) per batch.
// h = X@W (f32 WMMA), u/v/p/t scalars, per-column online softmax stats,
// h' = att@h (f32 WMMA), ELU. All intermediates live in LDS (~135KB).
// W and h are kept TRANSPOSED in LDS (padded stride) so WMMA B fragments are
// single aligned b64 LDS loads (no repack moves), and the u/v dot products
// are bank-conflict free.

#define B_    32
#define N_    256
#define FIN_  128
#define FOUT_ 64
#define WT_LD 130   // FIN_ + 2 pad: even (8B alignment) + bank-conflict free
#define HT_LD 258   // N_   + 2 pad: even (8B alignment) + bank-conflict free
#define ALPHA_ 0.2f
#define NEGINF_ (-1e12f)

typedef float v2f __attribute__((ext_vector_type(2)));
typedef float v8f __attribute__((ext_vector_type(8)));

struct G1buf { float Wt[FOUT_ * WT_LD]; float X[N_ * 32]; };  // 33.3KB + 32KB
union  ShU   { G1buf g1; float att[N_ * FOUT_]; };            // overlap: 66KB

__launch_bounds__(256)
__global__ void gat_fused_kernel(const float* __restrict__ inp,
                                 const int*   __restrict__ adj,
                                 const float* __restrict__ Wg,
                                 const float* __restrict__ ag,
                                 float*       __restrict__ out)
{
    __shared__ __align__(16) ShU   shu;                  // 66KB (Wt+X) / (att)
    __shared__ __align__(16) float sHt[FOUT_ * HT_LD];   // 66KB, h transposed
    __shared__ float sAvec[2 * FOUT_];
    __shared__ float sU[N_], sV[N_], sP[N_], sT[N_], sM[N_], sS[N_];

    const int b     = blockIdx.x;
    const int tid   = threadIdx.x;
    const int wave  = tid >> 5;
    const int lane  = tid & 31;
    const int lhalf = lane >> 4;   // 0 or 1
    const int lmod  = lane & 15;

    const float* Xb = inp + (size_t)b * N_ * FIN_;
    const float* Wb = Wg  + (size_t)b * FIN_ * FOUT_;
    const int*   Ab = adj + (size_t)b * N_ * N_;
    float*       Ob = out + (size_t)b * N_ * FOUT_;

    // ---- stage W (transposed) and a ----
    for (int idx = tid; idx < FIN_ * FOUT_; idx += 256) {
        int k = idx >> 6, n = idx & 63;                 // coalesced global read
        shu.g1.Wt[n * WT_LD + k] = Wb[idx];
    }
    for (int k = tid; k < 2 * FOUT_; k += 256) sAvec[k] = ag[k];

    // ---- GEMM1: h = X @ W  (V_WMMA_F32_16X16X4_F32) ----
    v8f acc[2][4];
    for (int tm = 0; tm < 2; ++tm)
        for (int nt = 0; nt < 4; ++nt)
            for (int e = 0; e < 8; ++e) acc[tm][nt][e] = 0.0f;

    for (int kc = 0; kc < FIN_ / 32; ++kc) {
        __syncthreads();                              // protect X buffer reuse
        for (int idx = tid; idx < N_ * 32; idx += 256) {
            int row = idx >> 5, col = idx & 31;
            shu.g1.X[idx] = Xb[row * FIN_ + kc * 32 + col];
        }
        __syncthreads();
        for (int k0 = 0; k0 < 32; k0 += 4) {
            // A fragments (16x4 f32): VGPR0 = K0|K2, VGPR1 = K1|K3 -> b64 load
            v2f afrag[2];
            for (int tm = 0; tm < 2; ++tm) {
                int arow = (wave * 2 + tm) * 16 + lmod;
                afrag[tm] = *(const v2f*)&shu.g1.X[arow * 32 + k0 + 2 * lhalf];
            }
            // B fragments from transposed W: contiguous b64 load
            int krow = kc * 32 + k0 + 2 * lhalf;
            v2f bfrag[4];
            for (int nt = 0; nt < 4; ++nt) {
                int n = nt * 16 + lmod;
                bfrag[nt] = *(const v2f*)&shu.g1.Wt[n * WT_LD + krow];
            }
            for (int tm = 0; tm < 2; ++tm)
                for (int nt = 0; nt < 4; ++nt)
                    acc[tm][nt] = __builtin_amdgcn_wmma_f32_16x16x4_f32(
                        false, afrag[tm], false, bfrag[nt], (short)0,
                        acc[tm][nt], false, false);
        }
    }

    // spill h tiles to LDS transposed (C/D: VGPR e -> row e + 8*lhalf, col lmod)
    for (int tm = 0; tm < 2; ++tm) {
        int mt = wave * 2 + tm;
        for (int nt = 0; nt < 4; ++nt)
            for (int e = 0; e < 8; ++e) {
                int row = mt * 16 + e + 8 * lhalf;      // node index r
                int col = nt * 16 + lmod;               // feature index f
                sHt[col * HT_LD + row] = acc[tm][nt][e];
            }
    }
    __syncthreads();

    // ---- u[r] = h[r].a1, v[r] = h[r].a2 ; p/t per the cat(dim=1) layout ----
    {
        int r = tid;
        float u = 0.f, v = 0.f;
        for (int f = 0; f < FOUT_; ++f) {
            float hv = sHt[f * HT_LD + r];              // conflict-free
            u += hv * sAvec[f];
            v += hv * sAvec[FOUT_ + f];
        }
        sU[r] = u; sV[r] = v;
    }
    __syncthreads();
    {
        int r = tid;
        float x = sU[r] + sV[r];
        sP[r] = (x > 0.f) ? x : ALPHA_ * x;                       // rows i<128
        float y = sU[(2 * r) & (N_ - 1)] + sV[(2 * r + 1) & (N_ - 1)];
        sT[r] = (y > 0.f) ? y : ALPHA_ * y;                       // rows i>=128
    }
    __syncthreads();

    // ---- per-column (axis=1) online softmax stats ----
    {
        int j = tid;
        int c = (j >= 128) ? 1 : 0;
        float tj = sT[j];
        float m = -3.4e38f, s = 0.f;
        for (int i = 0; i < N_; ++i) {
            float e  = (i < 128) ? sP[2 * i + c] : tj;
            float me = (Ab[i * N_ + j] > 0) ? e : NEGINF_;
            if (me > m) { s *= __expf(m - me); m = me; }
            s += __expf(me - m);
        }
        sM[j] = m; sS[j] = s;
    }
    __syncthreads();

    // ---- GEMM2: h' = att @ h, K chunked by 64 columns of j ----
    v8f acc2[2][4];
    for (int tm = 0; tm < 2; ++tm)
        for (int nt = 0; nt < 4; ++nt)
            for (int e = 0; e < 8; ++e) acc2[tm][nt][e] = 0.0f;

    for (int kc = 0; kc < 4; ++kc) {
        __syncthreads();                              // protect att buffer reuse
        for (int idx = tid; idx < N_ * 64; idx += 256) {
            int i  = idx >> 6;
            int jl = idx & 63;
            int j  = kc * 64 + jl;
            float e  = (i < 128) ? sP[2 * i + ((j >= 128) ? 1 : 0)] : sT[j];
            float me = (Ab[i * N_ + j] > 0) ? e : NEGINF_;
            shu.att[idx] = __expf(me - sM[j]) / sS[j];
        }
        __syncthreads();
        for (int k0 = 0; k0 < 64; k0 += 4) {
            v2f afrag[2];
            for (int tm = 0; tm < 2; ++tm) {
                int arow = (wave * 2 + tm) * 16 + lmod;
                afrag[tm] = *(const v2f*)&shu.att[arow * 64 + k0 + 2 * lhalf];
            }
            int jrow = kc * 64 + k0 + 2 * lhalf;
            v2f bfrag[4];
            for (int nt = 0; nt < 4; ++nt) {
                int n = nt * 16 + lmod;
                bfrag[nt] = *(const v2f*)&sHt[n * HT_LD + jrow];  // b64, aligned
            }
            for (int tm = 0; tm < 2; ++tm)
                for (int nt = 0; nt < 4; ++nt)
                    acc2[tm][nt] = __builtin_amdgcn_wmma_f32_16x16x4_f32(
                        false, afrag[tm], false, bfrag[nt], (short)0,
                        acc2[tm][nt], false, false);
        }
    }

    // ---- ELU + store ----
    for (int tm = 0; tm < 2; ++tm) {
        int mt = wave * 2 + tm;
        for (int nt = 0; nt < 4; ++nt)
            for (int e = 0; e < 8; ++e) {
                int row = mt * 16 + e + 8 * lhalf;
                int col = nt * 16 + lmod;
                float x = acc2[tm][nt][e];
                Ob[row * FOUT_ + col] = (x > 0.f) ? x : (__expf(x) - 1.0f);
            }
    }
}

extern "C" void kernel_launch(void* const* d_in, const int* in_sizes, int n_in,
                              void* d_out, int out_size, void* d_ws, size_t ws_size,
                              hipStream_t stream) {
    (void)in_sizes; (void)n_in; (void)out_size; (void)d_ws; (void)ws_size;
    const float* inp = (const float*)d_in[0];
    const int*   adj = (const int*)d_in[1];
    const float* W   = (const float*)d_in[2];
    const float* a   = (const float*)d_in[3];
    float* out = (float*)d_out;
    gat_fused_kernel<<<dim3(B_), dim3(256), 0, stream>>>(inp, adj, W, a, out);
}